// AnchorPointsBasedViewTransformation_69715909149240
// MI455X (gfx1250) — compile-verified
//
#include <hip/hip_runtime.h>

#define BEV_Y 200
#define BEV_X 176
#define BEV_Z 10

typedef float v2f __attribute__((ext_vector_type(2)));
typedef float v8f __attribute__((ext_vector_type(8)));

// ---------------------------------------------------------------------------
// Repack weights from (m, c, kh, kw) to (m, kk=kh*KS+kw, c) so the conv's
// hot loop walks contiguous channels for both A (weights) and B (im2col).
// ---------------------------------------------------------------------------
__global__ __launch_bounds__(256) void repack_weights_kernel(
    const float* __restrict__ w, float* __restrict__ wT, int Cout, int Cin, int KS2)
{
    const int i = blockIdx.x * blockDim.x + threadIdx.x;
    const int total = Cout * Cin * KS2;
    if (i >= total) return;
    const int kk = i % KS2;
    const int t  = i / KS2;
    const int c  = t % Cin;
    const int m  = t / Cin;
    wT[((size_t)m * KS2 + kk) * Cin + c] = w[i];
}

// ---------------------------------------------------------------------------
// Implicit-GEMM conv + folded-BN + ReLU using V_WMMA_F32_16X16X4_F32.
//   Each wave computes 64 output channels x 32 pixels:
//   4 M-tiles x 2 N-tiles = 8 C-fragments; per 4-k step: 4 A-loads + 4
//   B-loads feed 8 WMMAs (A reused 2x, B reused 4x).
//   GEMM: C[M=Cout, N=pixels] = Wt[M,K] * im2col[K,N], K ordered (kh,kw,c).
//   A-frag (16x4): lane 0-15 row M=lane;  VGPR j, half h -> K = k0 + j + 2h
//   B-frag (4x16): lane gives N;          VGPR j, half h -> K = k0 + j + 2h
//   C/D  (16x16):  VGPR j -> row M = j + 8*half, col N = lane&15
// ---------------------------------------------------------------------------
template<int KS, int STRIDE, int PAD, bool NHWC_OUT>
__global__ __launch_bounds__(256) void conv_wmma_kernel(
    const float* __restrict__ in, const float* __restrict__ wT,
    const float* __restrict__ scale, const float* __restrict__ shift,
    float* __restrict__ out,
    int B, int Cin, int Hin, int Win, int Cout, int Hout, int Wout)
{
    const int lane = threadIdx.x & 31;
    const int lid  = lane & 15;
    const int half = lane >> 4;
    const int groupsM = Cout >> 6;                 // 64 out-channels per wave
    const int npix    = B * Hout * Wout;
    const int tilesN  = (npix + 31) >> 5;          // 32 pixels per wave
    const int wave = blockIdx.x * (blockDim.x >> 5) + (threadIdx.x >> 5);
    if (wave >= groupsM * tilesN) return;
    const int tm = wave % groupsM;
    const int tn = wave / groupsM;

    const int K = Cin * KS * KS;
    const size_t HW = (size_t)Hin * Win;

    // ---- two pixel tiles (N-tiles) --------------------------------------
    const int nn1 = (tn << 5) + lid;               // always < npix
    const int nn2 = nn1 + 16;
    const bool tile2ok = ((tn << 5) + 16) < npix;  // wave-uniform
    const int nn2c = tile2ok ? nn2 : nn1;

    int ox1 = nn1 % Wout, tq1 = nn1 / Wout;
    int oy1 = tq1 % Hout, bb1 = tq1 / Hout;
    int ox2 = nn2c % Wout, tq2 = nn2c / Wout;
    int oy2 = tq2 % Hout, bb2 = tq2 / Hout;

    const float* __restrict__ inb1 = in + (size_t)bb1 * Cin * HW;
    const float* __restrict__ inb2 = in + (size_t)bb2 * Cin * HW;
    const int iy10 = oy1 * STRIDE - PAD, ix10 = ox1 * STRIDE - PAD;
    const int iy20 = oy2 * STRIDE - PAD, ix20 = ox2 * STRIDE - PAD;

    // ---- 4 A-row pointers; k-major layout wT[m][kk*Cin+c] ---------------
    const float* __restrict__ wr[4];
    wr[0] = wT + (size_t)((tm << 6) + lid) * K + (half << 1);
    wr[1] = wr[0] + (size_t)16 * K;
    wr[2] = wr[0] + (size_t)32 * K;
    wr[3] = wr[0] + (size_t)48 * K;

    v8f acc[2][4] = {};

#pragma unroll
    for (int kk = 0; kk < KS * KS; ++kk) {
        const int kh = kk / KS;
        const int kw = kk - kh * KS;
        const int iy1 = iy10 + kh, ix1 = ix10 + kw;
        const int iy2 = iy20 + kh, ix2 = ix20 + kw;
        const bool valid1 = ((unsigned)iy1 < (unsigned)Hin) && ((unsigned)ix1 < (unsigned)Win);
        const bool valid2 = tile2ok &&
                            ((unsigned)iy2 < (unsigned)Hin) && ((unsigned)ix2 < (unsigned)Win);
        // clamped pointers + cndmask keep the loop branch-free (EXEC all-1s)
        const float* __restrict__ bp1 =
            inb1 + (valid1 ? ((size_t)iy1 * Win + ix1) : (size_t)0) + (size_t)(half << 1) * HW;
        const float* __restrict__ bp2 =
            inb2 + (valid2 ? ((size_t)iy2 * Win + ix2) : (size_t)0) + (size_t)(half << 1) * HW;
        const float* __restrict__ a0 = wr[0] + (size_t)kk * Cin;
        const float* __restrict__ a1 = wr[1] + (size_t)kk * Cin;
        const float* __restrict__ a2 = wr[2] + (size_t)kk * Cin;
        const float* __restrict__ a3 = wr[3] + (size_t)kk * Cin;

#pragma unroll 4
        for (int c0 = 0; c0 < Cin; c0 += 4) {
            v2f af[4];
            af[0] = *(const v2f*)(a0 + c0);
            af[1] = *(const v2f*)(a1 + c0);
            af[2] = *(const v2f*)(a2 + c0);
            af[3] = *(const v2f*)(a3 + c0);
            const float t0 = bp1[(size_t)c0 * HW];
            const float t1 = bp1[(size_t)(c0 + 1) * HW];
            const float u0 = bp2[(size_t)c0 * HW];
            const float u1 = bp2[(size_t)(c0 + 1) * HW];
            v2f bf0, bf1;
            bf0.x = valid1 ? t0 : 0.0f;
            bf0.y = valid1 ? t1 : 0.0f;
            bf1.x = valid2 ? u0 : 0.0f;
            bf1.y = valid2 ? u1 : 0.0f;
#pragma unroll
            for (int f = 0; f < 4; ++f) {
                acc[0][f] = __builtin_amdgcn_wmma_f32_16x16x4_f32(
                    false, af[f], false, bf0, (short)0, acc[0][f], false, false);
                acc[1][f] = __builtin_amdgcn_wmma_f32_16x16x4_f32(
                    false, af[f], false, bf1, (short)0, acc[1][f], false, false);
            }
        }
    }

    // ---- epilogue: folded BN + ReLU, coalesced stores -------------------
#pragma unroll
    for (int t = 0; t < 2; ++t) {
        if (t == 1 && !tile2ok) break;             // wave-uniform
        const int bb = t ? bb2 : bb1;
        const int oy = t ? oy2 : oy1;
        const int ox = t ? ox2 : ox1;
#pragma unroll
        for (int f = 0; f < 4; ++f) {
#pragma unroll
            for (int j = 0; j < 8; ++j) {
                const int mrow = (tm << 6) + (f << 4) + j + (half << 3);  // out ch
                float v = acc[t][f][j] * scale[mrow] + shift[mrow];
                v = fmaxf(v, 0.0f);
                if (NHWC_OUT)
                    out[(((size_t)bb * Hout + oy) * Wout + ox) * Cout + mrow] = v;
                else
                    out[(((size_t)bb * Cout + mrow) * Hout + oy) * Wout + ox] = v;
            }
        }
    }
}

// ---------------------------------------------------------------------------
// pemask(b,y,x,c) = (1/Z) * sum_z sigmoid(MLP(pos(y,x,z)))_c * mask(b,y,x,z)
// MLP weights staged in LDS; one thread per (b,y,x).
// ---------------------------------------------------------------------------
__global__ __launch_bounds__(64) void pemask_kernel(
    const float* __restrict__ mask,
    const float* __restrict__ w1, const float* __restrict__ b1,
    const float* __restrict__ w2, const float* __restrict__ b2,
    const float* __restrict__ w3, const float* __restrict__ b3,
    float* __restrict__ pem, int B)
{
    __shared__ float s_w1[48], s_b1[16], s_w2[512], s_b2[32], s_w3[2048], s_b3[64];
    const int tid = threadIdx.x;
    if (tid < 48) s_w1[tid] = w1[tid];
    if (tid < 16) s_b1[tid] = b1[tid];
    for (int i = tid; i < 512; i += 64) s_w2[i] = w2[i];
    if (tid < 32) s_b2[tid] = b2[tid];
    for (int i = tid; i < 2048; i += 64) s_w3[i] = w3[i];
    s_b3[tid] = b3[tid];
    __syncthreads();

    const int idx = blockIdx.x * 64 + tid;
    if (idx >= B * BEV_Y * BEV_X) return;
    const int x = idx % BEV_X;
    const int y = (idx / BEV_X) % BEV_Y;
    const float py = (float)y / (float)BEV_Y;
    const float px = (float)x / (float)BEV_X;

    float acc[64];
#pragma unroll
    for (int c = 0; c < 64; ++c) acc[c] = 0.0f;

    const float* mp = mask + (size_t)idx * BEV_Z;
    for (int z = 0; z < BEV_Z; ++z) {
        const float pz = (float)z / (float)BEV_Z;
        float h1[16];
#pragma unroll
        for (int k = 0; k < 16; ++k) {
            float v = py * s_w1[k] + px * s_w1[16 + k] + pz * s_w1[32 + k] + s_b1[k];
            h1[k] = fmaxf(v, 0.0f);
        }
        float h2[32];
#pragma unroll
        for (int k = 0; k < 32; ++k) {
            float v = s_b2[k];
#pragma unroll
            for (int j = 0; j < 16; ++j) v += h1[j] * s_w2[j * 32 + k];
            h2[k] = fmaxf(v, 0.0f);
        }
        const float mz = mp[z];
#pragma unroll
        for (int c = 0; c < 64; ++c) {
            float v = s_b3[c];
#pragma unroll
            for (int j = 0; j < 32; ++j) v += h2[j] * s_w3[j * 64 + c];
            v = 1.0f / (1.0f + __expf(-v));
            acc[c] += v * mz;
        }
    }
    float* op = pem + (size_t)idx * 64;
#pragma unroll
    for (int c = 0; c < 64; ++c) op[c] = acc[c] * (1.0f / BEV_Z);
}

// ---------------------------------------------------------------------------
// Gather adapted NHWC features at (v,u), masked mean over z, add pemask,
// write bevf in NCHW (B,192,Y,X). One thread = (b,lev,cgroup-of-4,y,x).
// ---------------------------------------------------------------------------
__global__ __launch_bounds__(256) void gather_kernel(
    const float* __restrict__ f0, const float* __restrict__ f1, const float* __restrict__ f2,
    const int* __restrict__ uv, const float* __restrict__ mask,
    const float* __restrict__ pem, float* __restrict__ bevf, int B)
{
    int t = blockIdx.x * blockDim.x + threadIdx.x;
    const int total = B * 3 * 16 * BEV_Y * BEV_X;
    if (t >= total) return;
    const int x   = t % BEV_X;  t /= BEV_X;
    const int y   = t % BEV_Y;  t /= BEV_Y;
    const int cg  = t % 16;     t /= 16;
    const int lev = t % 3;
    const int b   = t / 3;

    const float* f; int H, W;
    if (lev == 0)      { f = f0; H = 96; W = 320; }
    else if (lev == 1) { f = f1; H = 48; W = 160; }
    else               { f = f2; H = 24; W = 80;  }

    const size_t pbase = ((size_t)b * BEV_Y + y) * BEV_X + x;
    const int*   uvp = uv   + pbase * BEV_Z * 2;
    const float* mp  = mask + pbase * BEV_Z;

    float a0 = 0.f, a1 = 0.f, a2 = 0.f, a3 = 0.f;
#pragma unroll
    for (int z = 0; z < BEV_Z; ++z) {
        const int u = (uvp[2 * z]     >> 2) >> lev;   // uv//4 then //2^lev
        const int v = (uvp[2 * z + 1] >> 2) >> lev;
        const float4 g = *(const float4*)(f + (((size_t)b * H + v) * W + u) * 64 + (cg << 2));
        const float mz = mp[z];
        a0 += g.x * mz; a1 += g.y * mz; a2 += g.z * mz; a3 += g.w * mz;
    }
    const float4 pe = *(const float4*)(pem + pbase * 64 + (cg << 2));
    const int c0 = lev * 64 + (cg << 2);
    const size_t plane = (size_t)BEV_Y * BEV_X;
    float* op = bevf + ((size_t)b * 192 + c0) * plane + (size_t)y * BEV_X + x;
    op[0]         = a0 * (1.0f / BEV_Z) + pe.x;
    op[plane]     = a1 * (1.0f / BEV_Z) + pe.y;
    op[2 * plane] = a2 * (1.0f / BEV_Z) + pe.z;
    op[3 * plane] = a3 * (1.0f / BEV_Z) + pe.w;
}

// ---------------------------------------------------------------------------
// out = relu(p0 + nearest_up2(p1) + nearest_up4(p2)); p0 already in out.
// ---------------------------------------------------------------------------
__global__ __launch_bounds__(256) void fuse_kernel(
    const float* __restrict__ p1, const float* __restrict__ p2,
    float* __restrict__ out, int B)
{
    const int i = blockIdx.x * blockDim.x + threadIdx.x;
    const int total = B * 64 * BEV_Y * BEV_X;
    if (i >= total) return;
    const int x = i % BEV_X;
    int t = i / BEV_X;
    const int y = t % BEV_Y; t /= BEV_Y;
    const int c = t % 64;
    const int b = t / 64;
    float v = out[i];
    v += p1[(((size_t)b * 64 + c) * 100 + (y >> 1)) * 88 + (x >> 1)];
    v += p2[(((size_t)b * 64 + c) * 50  + (y >> 2)) * 44 + (x >> 2)];
    out[i] = fmaxf(v, 0.0f);
}

// ---------------------------------------------------------------------------

static inline int conv_blocks(int Cout, int npix)
{
    const int waves = (Cout >> 6) * ((npix + 31) >> 5);
    return (waves + 7) / 8;   // 8 waves (256 threads) per block
}

extern "C" void kernel_launch(void* const* d_in, const int* in_sizes, int n_in,
                              void* d_out, int out_size, void* d_ws, size_t ws_size,
                              hipStream_t stream)
{
    const float* feat0 = (const float*)d_in[0];
    const float* feat1 = (const float*)d_in[1];
    const float* feat2 = (const float*)d_in[2];
    // d_in[3] = anchor_points_xyz (unused by the reference)
    const int*   uv    = (const int*)d_in[4];
    const float* mask  = (const float*)d_in[5];
    const float* ada_w[3] = {(const float*)d_in[6],  (const float*)d_in[9],  (const float*)d_in[12]};
    const float* ada_s[3] = {(const float*)d_in[7],  (const float*)d_in[10], (const float*)d_in[13]};
    const float* ada_b[3] = {(const float*)d_in[8],  (const float*)d_in[11], (const float*)d_in[14]};
    const float* pe_w1 = (const float*)d_in[15];
    const float* pe_b1 = (const float*)d_in[16];
    const float* pe_w2 = (const float*)d_in[17];
    const float* pe_b2 = (const float*)d_in[18];
    const float* pe_w3 = (const float*)d_in[19];
    const float* pe_b3 = (const float*)d_in[20];
    const float* cin_w = (const float*)d_in[21];
    const float* cin_s = (const float*)d_in[22];
    const float* cin_b = (const float*)d_in[23];
    const float* cd_w0 = (const float*)d_in[24];
    const float* cd_s0 = (const float*)d_in[25];
    const float* cd_b0 = (const float*)d_in[26];
    const float* cd_w1 = (const float*)d_in[27];
    const float* cd_s1 = (const float*)d_in[28];
    const float* cd_b1 = (const float*)d_in[29];
    const float* up_w0 = (const float*)d_in[30];
    const float* up_s0 = (const float*)d_in[31];
    const float* up_b0 = (const float*)d_in[32];
    const float* up_w1 = (const float*)d_in[33];
    const float* up_s1 = (const float*)d_in[34];
    const float* up_b1 = (const float*)d_in[35];
    const float* up_w2 = (const float*)d_in[36];
    const float* up_s2 = (const float*)d_in[37];
    const float* up_b2 = (const float*)d_in[38];

    const int B = in_sizes[0] / (64 * 96 * 320);
    const size_t plane = (size_t)BEV_Y * BEV_X;

    // Workspace carve (floats), with lifetime-based region reuse:
    //   region A: adapted feats (NHWC) -> reused by l0 after gather
    //   region B: pemask
    //   region C: bevf (NCHW)          -> reused by l1/l2/p1/p2 after conv_in
    //   region D: repacked weights (persistent for the whole launch)
    float* ws = (float*)d_ws;
    const size_t szF0 = (size_t)B * 96 * 320 * 64;
    const size_t szF1 = (size_t)B * 48 * 160 * 64;
    const size_t szF2 = (size_t)B * 24 * 80  * 64;
    float* f0a = ws;
    float* f1a = f0a + szF0;
    float* f2a = f1a + szF1;
    float* regionA_end = f2a + szF2;
    float* pem  = regionA_end;                          // B*Y*X*64
    float* bevf = pem + (size_t)B * plane * 64;         // B*192*Y*X
    float* l0 = ws;                                     // reuse region A
    float* l1 = bevf;                                   // reuse region C
    float* l2 = l1 + (size_t)B * 128 * 100 * 88;
    float* p1 = l2 + (size_t)B * 256 * 50 * 44;
    float* p2 = p1 + (size_t)B * 64 * 100 * 88;
    float* p0 = (float*)d_out;

    // region D: repacked weights
    float* wtp = bevf + (size_t)B * 192 * plane;
    const int wcfg[9][3] = {   // {Cout, Cin, KS2}
        {64, 64, 9}, {64, 128, 9}, {64, 256, 9},       // ada0..2
        {64, 192, 9},                                  // cin
        {128, 64, 16}, {256, 128, 16},                 // cd0, cd1
        {64, 64, 9}, {64, 128, 9}, {64, 256, 9}        // up0..2
    };
    const float* wsrc[9] = {ada_w[0], ada_w[1], ada_w[2], cin_w, cd_w0, cd_w1,
                            up_w0, up_w1, up_w2};
    float* wdst[9];
    {
        float* p = wtp;
        for (int i = 0; i < 9; ++i) {
            wdst[i] = p;
            const int total = wcfg[i][0] * wcfg[i][1] * wcfg[i][2];
            repack_weights_kernel<<<(total + 255) / 256, 256, 0, stream>>>(
                wsrc[i], wdst[i], wcfg[i][0], wcfg[i][1], wcfg[i][2]);
            p += total;
        }
    }

    // 1) adaptation convs (3x3, s1, p1), NCHW in -> NHWC out for fast gather
    conv_wmma_kernel<3,1,1,true><<<conv_blocks(64, B*96*320), 256, 0, stream>>>(
        feat0, wdst[0], ada_s[0], ada_b[0], f0a, B, 64, 96, 320, 64, 96, 320);
    conv_wmma_kernel<3,1,1,true><<<conv_blocks(64, B*48*160), 256, 0, stream>>>(
        feat1, wdst[1], ada_s[1], ada_b[1], f1a, B, 128, 48, 160, 64, 48, 160);
    conv_wmma_kernel<3,1,1,true><<<conv_blocks(64, B*24*80), 256, 0, stream>>>(
        feat2, wdst[2], ada_s[2], ada_b[2], f2a, B, 256, 24, 80, 64, 24, 80);

    // 2) pos-embed MLP folded with mask: pemask = mean_z sigmoid(MLP)*mask
    pemask_kernel<<<(B * BEV_Y * BEV_X + 63) / 64, 64, 0, stream>>>(
        mask, pe_w1, pe_b1, pe_w2, pe_b2, pe_w3, pe_b3, pem, B);

    // 3) gather + masked mean over depth bins -> bevf (B,192,Y,X)
    {
        const int total = B * 3 * 16 * BEV_Y * BEV_X;
        gather_kernel<<<(total + 255) / 256, 256, 0, stream>>>(
            f0a, f1a, f2a, uv, mask, pem, bevf, B);
    }

    // 4) conv_in 192->64 (3x3)
    conv_wmma_kernel<3,1,1,false><<<conv_blocks(64, B*BEV_Y*BEV_X), 256, 0, stream>>>(
        bevf, wdst[3], cin_s, cin_b, l0, B, 192, BEV_Y, BEV_X, 64, BEV_Y, BEV_X);

    // 5) down convs (4x4, s2, p1)
    conv_wmma_kernel<4,2,1,false><<<conv_blocks(128, B*100*88), 256, 0, stream>>>(
        l0, wdst[4], cd_s0, cd_b0, l1, B, 64, BEV_Y, BEV_X, 128, 100, 88);
    conv_wmma_kernel<4,2,1,false><<<conv_blocks(256, B*50*44), 256, 0, stream>>>(
        l1, wdst[5], cd_s1, cd_b1, l2, B, 128, 100, 88, 256, 50, 44);

    // 6) per-level 3x3 projections to 64ch (p0 goes straight into d_out)
    conv_wmma_kernel<3,1,1,false><<<conv_blocks(64, B*BEV_Y*BEV_X), 256, 0, stream>>>(
        l0, wdst[6], up_s0, up_b0, p0, B, 64, BEV_Y, BEV_X, 64, BEV_Y, BEV_X);
    conv_wmma_kernel<3,1,1,false><<<conv_blocks(64, B*100*88), 256, 0, stream>>>(
        l1, wdst[7], up_s1, up_b1, p1, B, 128, 100, 88, 64, 100, 88);
    conv_wmma_kernel<3,1,1,false><<<conv_blocks(64, B*50*44), 256, 0, stream>>>(
        l2, wdst[8], up_s2, up_b2, p2, B, 256, 50, 44, 64, 50, 44);

    // 7) fuse: out = relu(p0 + up2(p1) + up4(p2))
    {
        const int total = B * 64 * BEV_Y * BEV_X;
        fuse_kernel<<<(total + 255) / 256, 256, 0, stream>>>(p1, p2, (float*)d_out, B);
    }
}